// RelativePositionEncoding_1468878815797
// MI455X (gfx1250) — compile-verified
//
#include <hip/hip_runtime.h>

// RelativePositionEncoding for MI455X (gfx1250).
// Store-bandwidth-bound gather kernel: 512 MB f32 output at 23.3 TB/s => ~22us floor.
// W table (139x128 f32 = 71KB) is transpose-staged into LDS via
// GLOBAL_LOAD_ASYNC_TO_LDS_B32 (per-lane global->per-lane LDS scatter, ASYNCcnt),
// hot loop is wave-uniform index math + conflict-free ds_load_b128 gathers +
// nontemporal global_store_b128.

#define RP_RMAX 32
#define RP_SMAX 2
#define RP_CZ 128
#define RP_FEATS 139          // 4*R_MAX + 2*S_MAX + 7
#define RP_LDSROW 132         // padded row pitch (floats): (4f+c) banking, 16B-aligned rows
#define RP_THREADS 256

typedef float rp_float4 __attribute__((ext_vector_type(4)));

typedef __attribute__((address_space(1))) int rp_gint;
typedef __attribute__((address_space(3))) int rp_lint;

#if defined(__has_builtin)
#  if __has_builtin(__builtin_amdgcn_global_load_async_to_lds_b32)
#    define RP_HAVE_ASYNC 1
#  endif
#endif

__global__ __launch_bounds__(RP_THREADS) void
relpos_gather_kernel(const float* __restrict__ W,     // [C_Z, IN_FEATS] row-major
                     const int* __restrict__ asym,
                     const int* __restrict__ resi,
                     const int* __restrict__ ent,
                     const int* __restrict__ sym,
                     const int* __restrict__ tok,
                     float* __restrict__ out,          // [N, N, C_Z]
                     int N)
{
    __shared__ __align__(16) float wt[RP_FEATS * RP_LDSROW]; // Wt[f][c], pitch 132

    const int tid = threadIdx.x;

    // Pull the small index arrays into WGP$ up front (gfx1250 global_prefetch_b8).
    {
        int pi = (tid * 4) & (N - 1);
        __builtin_prefetch(asym + pi, 0, 0);
        __builtin_prefetch(resi + pi, 0, 0);
        __builtin_prefetch(ent  + pi, 0, 0);
        __builtin_prefetch(sym  + pi, 0, 0);
        __builtin_prefetch(tok  + pi, 0, 0);
    }

    // ---- Stage W transposed into LDS: wt[f*132 + c] = W[c*139 + f] ----
    // Global side fully coalesced (idx walks W linearly); LDS side is a
    // per-lane scatter -> GLOBAL_LOAD_ASYNC_TO_LDS_B32 (ASYNCcnt path).
    for (int idx = tid; idx < RP_CZ * RP_FEATS; idx += RP_THREADS) {
        int c = idx / RP_FEATS;
        int f = idx - c * RP_FEATS;
#if defined(RP_HAVE_ASYNC)
        int* gsrc = (int*)(const_cast<float*>(W) + idx);
        int* ldst = (int*)(&wt[f * RP_LDSROW + c]);
        __builtin_amdgcn_global_load_async_to_lds_b32(
            (rp_gint*)gsrc, (rp_lint*)ldst, 0, 0);
#else
        wt[f * RP_LDSROW + c] = W[idx];
#endif
    }
#if defined(RP_HAVE_ASYNC)
#  if defined(__has_builtin) && __has_builtin(__builtin_amdgcn_s_wait_asynccnt)
    __builtin_amdgcn_s_wait_asynccnt(0);
#  else
    asm volatile("s_wait_asynccnt 0" ::: "memory");
#  endif
#endif
    __syncthreads();

    const int i    = blockIdx.x;       // output row
    const int lane = tid & 31;         // channel quad: channels [4*lane, 4*lane+3]
    const int wv   = tid >> 5;         // 8 waves -> 8 interleaved j's

    // Row-i indices: wave-uniform -> SALU/s_load.
    const int ai = asym[i], ri = resi[i], ei = ent[i], si = sym[i], ti = tok[i];

    float* __restrict__ orow = out + (size_t)i * (size_t)N * RP_CZ;

    for (int j = wv; j < N; j += 8) {
        const int aj = asym[j], rj = resi[j], ej = ent[j], sj = sym[j], tj = tok[j];

        const bool same_chain = (ai == aj);
        const bool same_res   = (ri == rj);
        const bool same_ent   = (ei == ej);

        int dres = ri - rj + RP_RMAX;
        dres = min(max(dres, 0), 2 * RP_RMAX);
        const int r0 = same_chain ? dres : (2 * RP_RMAX + 1);            // [0,65]

        int dtok = ti - tj + RP_RMAX;
        dtok = min(max(dtok, 0), 2 * RP_RMAX);
        const int r1 = 2 * (RP_RMAX + 1) +
                       ((same_chain && same_res) ? dtok : (2 * RP_RMAX + 1)); // [66,131]

        int dch = si - sj + RP_SMAX;
        dch = min(max(dch, 0), 2 * RP_SMAX);
        const int r3 = (4 * RP_RMAX + 5) + (same_ent ? dch : (2 * RP_SMAX + 1)); // [133,138]

        const float m = same_ent ? 1.0f : 0.0f;

        // Conflict-free ds_load_b128 gathers (lane l -> words 4l..4l+3 of a row).
        const rp_float4 v0 = *(const rp_float4*)&wt[r0 * RP_LDSROW + lane * 4];
        const rp_float4 v1 = *(const rp_float4*)&wt[r1 * RP_LDSROW + lane * 4];
        const rp_float4 v2 = *(const rp_float4*)&wt[(4 * RP_RMAX + 4) * RP_LDSROW + lane * 4]; // row 132
        const rp_float4 v3 = *(const rp_float4*)&wt[r3 * RP_LDSROW + lane * 4];

        // Match reference FP order: ((res + tok) + ent*W132) + ch  (m in {0,1} exact)
        rp_float4 acc = v0 + v1;
        acc += m * v2;
        acc += v3;

        // Streaming (non-temporal) 16B store: 512B per wave per pair -> pure HBM stream.
        __builtin_nontemporal_store(
            acc, (rp_float4*)(orow + (size_t)j * RP_CZ + lane * 4));
    }
}

extern "C" void kernel_launch(void* const* d_in, const int* in_sizes, int n_in,
                              void* d_out, int out_size, void* d_ws, size_t ws_size,
                              hipStream_t stream) {
    (void)n_in; (void)out_size; (void)d_ws; (void)ws_size;
    const float* W  = (const float*)d_in[0];
    const int* asym = (const int*)d_in[1];
    const int* resi = (const int*)d_in[2];
    const int* ent  = (const int*)d_in[3];
    const int* sym  = (const int*)d_in[4];
    const int* tok  = (const int*)d_in[5];
    const int N = in_sizes[1];   // 1024 tokens

    relpos_gather_kernel<<<N, RP_THREADS, 0, stream>>>(
        W, asym, resi, ent, sym, tok, (float*)d_out, N);
}